// GIN_60078002536570
// MI455X (gfx1250) — compile-verified
//
#include <hip/hip_runtime.h>

#define N_NODES 40000
#define N_EDGES 640000
#define DIM 128

typedef __attribute__((ext_vector_type(2))) float v2f;
typedef __attribute__((ext_vector_type(8))) float v8f;

__device__ __forceinline__ int lower_bound_i32(const int* __restrict__ arr, int n, int key) {
  int lo = 0, hi = n;
  while (lo < hi) {
    int mid = (lo + hi) >> 1;
    if (arr[mid] < key) lo = mid + 1; else hi = mid;
  }
  return lo;
}

// One block per node, 128 threads (one feature column each).
// dst_edges is sorted, so each node owns a contiguous edge range: deterministic
// sequential segment-sum, no float atomics. src[e] is block-uniform -> s_load.
__global__ __launch_bounds__(DIM) void gin_aggregate(
    const int* __restrict__ src, const int* __restrict__ dst,
    const float* __restrict__ h, float* __restrict__ agg) {
  const int node = blockIdx.x;
  const int col  = threadIdx.x;
  // Redundant uniform binary search per thread (scalarizes; ~20 iters).
  const int e0 = lower_bound_i32(dst, N_EDGES, node);
  const int e1 = lower_bound_i32(dst, N_EDGES, node + 1);
  float acc = 0.0f;
  for (int e = e0; e < e1; ++e) {
    const int s = src[e];
    acc += h[(size_t)s * DIM + col];
  }
  agg[(size_t)node * DIM + col] = acc;
}

// out[M=16 tile] = agg[16 x 128] @ W[128 x DOUT], fp32 WMMA 16x16x4, K-step 4.
// Block: (DOUT/16) waves; wave w computes the 16x16 output tile at columns
// [16w, 16w+16). Fragment layouts per CDNA5 ISA 7.12.2:
//   A (16x4): lanes 0-15 -> K=0(v0),K=1(v1); lanes 16-31 -> K=2(v0),K=3(v1)
//   B (4x16): lanes 0-15 -> rows K=0,1;      lanes 16-31 -> rows K=2,3
//   C/D: VGPR r, lanes 0-15 -> M=r, N=lane; lanes 16-31 -> M=r+8, N=lane-16
template <int DOUT>
__global__ __launch_bounds__(32 * (DOUT / 16)) void gin_gemm_wmma(
    const float* __restrict__ agg, const float* __restrict__ W,
    float* __restrict__ out) {
  const int wave  = threadIdx.x >> 5;       // output N-tile index
  const int lane  = threadIdx.x & 31;
  const int mbase = blockIdx.x * 16;
  const int arow  = mbase + (lane & 15);    // A-fragment row for this lane
  const int kofs  = (lane >> 4) << 1;       // 0 (lanes 0-15) or 2 (lanes 16-31)
  const int col   = wave * 16 + (lane & 15);

  const float* __restrict__ aptr = agg + (size_t)arow * DIM;

  v8f acc = {};
#pragma unroll 4
  for (int k = 0; k < DIM; k += 4) {
    v2f a, b;
    a.x = aptr[k + kofs];
    a.y = aptr[k + kofs + 1];
    b.x = W[(size_t)(k + kofs) * DOUT + col];
    b.y = W[(size_t)(k + kofs + 1) * DOUT + col];
    acc = __builtin_amdgcn_wmma_f32_16x16x4_f32(
        /*neg_a=*/false, a, /*neg_b=*/false, b,
        /*c_mod=*/(short)0, acc, /*reuse_a=*/false, /*reuse_b=*/false);
  }

  const int rbase = mbase + ((lane >> 4) << 3);  // +0 or +8
#pragma unroll
  for (int r = 0; r < 8; ++r) {
    out[(size_t)(rbase + r) * DOUT + col] = acc[r];
  }
}

extern "C" void kernel_launch(void* const* d_in, const int* in_sizes, int n_in,
                              void* d_out, int out_size, void* d_ws, size_t ws_size,
                              hipStream_t stream) {
  const int*   src  = (const int*)d_in[0];
  const int*   dst  = (const int*)d_in[1];
  const float* feat = (const float*)d_in[2];
  const float* W1   = (const float*)d_in[3];
  const float* W2   = (const float*)d_in[4];
  const float* W3   = (const float*)d_in[5];
  float*       out  = (float*)d_out;

  float* hbuf   = (float*)d_ws;                          // N*128 f32 (20.5 MB)
  float* aggbuf = hbuf + (size_t)N_NODES * DIM;          // N*128 f32 (20.5 MB)

  const dim3 aggGrid(N_NODES);
  const dim3 aggBlk(DIM);
  const dim3 gemmGrid(N_NODES / 16);                     // 2500 16-row tiles

  // Layer 1: agg = segsum(feat); h = agg @ W1
  gin_aggregate<<<aggGrid, aggBlk, 0, stream>>>(src, dst, feat, aggbuf);
  gin_gemm_wmma<128><<<gemmGrid, 256, 0, stream>>>(aggbuf, W1, hbuf);
  // Layer 2: agg = segsum(h); h = agg @ W2
  gin_aggregate<<<aggGrid, aggBlk, 0, stream>>>(src, dst, hbuf, aggbuf);
  gin_gemm_wmma<128><<<gemmGrid, 256, 0, stream>>>(aggbuf, W2, hbuf);
  // Layer 3: agg = segsum(h); out = agg @ W3 (DOUT=64)
  gin_aggregate<<<aggGrid, aggBlk, 0, stream>>>(src, dst, hbuf, aggbuf);
  gin_gemm_wmma<64><<<gemmGrid, 128, 0, stream>>>(aggbuf, W3, out);
}